// AdaptivePooling_33509334844022
// MI455X (gfx1250) — compile-verified
//
#include <hip/hip_runtime.h>
#include <hip/hip_bf16.h>
#include <math.h>
#include <stdint.h>

// ---------------------------------------------------------------------------
// AdaptivePooling: windowed (mean,max,logvar) pooling + batched projection GEMM
//
// Stage 1 (pool_kernel):  x[64,256,4096] -> pooled[64,256,512] (col 511 = pad)
//   one 256-thread block per (b,d) row; row staged in 16KB LDS via the CDNA5
//   async global->LDS path (global_load_async_to_lds_b128, ASYNCcnt).
// Stage 2 (gemm_kernel):  out[b] = proj_w @ pooled[b] + bias, fp32 WMMA.
//   one wave32 per 64x16 output block (4 M-tiles sharing one B fragment),
//   K=256 in steps of 4 via v_wmma_f32_16x16x4_f32 (exact fp32 — GEMM is
//   4.3 GFLOP, trivially below any ceiling; the op is HBM-bound).
// ---------------------------------------------------------------------------

#define TLEN 4096
#define DDIM 256
#define BATCH 64
#define NWIN 511
#define NPAD 512

typedef __attribute__((ext_vector_type(2))) float v2f;
typedef __attribute__((ext_vector_type(8))) float v8f;

__global__ void __launch_bounds__(256)
pool_kernel(const float* __restrict__ x,
            const float* __restrict__ pw,
            float* __restrict__ pooled) {
    __shared__ float row[TLEN];
    const int bd  = blockIdx.x;              // 0 .. 64*256-1  == (b,d) row id
    const int tid = threadIdx.x;             // 0 .. 255

    // ---- async-copy the whole row into LDS (b128 per lane, 4 rounds) ----
    // Direct global->LDS DMA path: no VGPR round-trip, tracked by ASYNCcnt.
    {
        const float* gsrc = x + (size_t)bd * TLEN + tid * 4;       // 16B/lane
        unsigned     ldst = (unsigned)(uintptr_t)(&row[tid * 4]);  // LDS byte addr
#pragma unroll
        for (int j = 0; j < 4; ++j) {
            asm volatile("global_load_async_to_lds_b128 %0, %1, off"
                         :
                         : "v"(ldst + j * 4096), "v"(gsrc + j * 1024)
                         : "memory");
        }
    }

    // ---- softmax of the 3 pool weights while the DMA is in flight ----
    float w0 = pw[0], w1 = pw[1], w2 = pw[2];
    float m  = fmaxf(w0, fmaxf(w1, w2));
    float e0 = __expf(w0 - m), e1 = __expf(w1 - m), e2 = __expf(w2 - m);
    float inv = 1.0f / (e0 + e1 + e2);
    w0 = e0 * inv; w1 = e1 * inv; w2 = e2 * inv;

    asm volatile("s_wait_asynccnt 0x0" ::: "memory");
    __syncthreads();

    float* orow = pooled + (size_t)bd * NPAD;

    // 511 windows + 1 pad column, 2 rounds over 256 threads
    for (int i = tid; i < NPAD; i += 256) {
        if (i < NWIN) {
            const int s = i * 8;
            float sum = 0.0f, sq = 0.0f, mx = -INFINITY;
#pragma unroll
            for (int j = 0; j < 16; ++j) {
                float v = row[s + j];
                sum += v;
                sq  += v * v;
                mx   = fmaxf(mx, v);
            }
            float mean = sum * (1.0f / 16.0f);
            float var  = (sq - 16.0f * mean * mean) * (1.0f / 15.0f);
            var = fminf(fmaxf(var, 1e-6f), 1e6f);
            float lv = __logf(var);
            orow[i] = w0 * mean + w1 * mx + w2 * lv;
        } else {
            orow[i] = 0.0f;  // pad column so the last N-tile is benign
        }
    }
}

// One wave32 computes a 64x16 block (4 stacked 16x16 WMMA tiles) of
// out[b] = W(256x256) @ pooled[b](256x512).  B fragment is loaded once per
// K-step and reused by all 4 tiles (1.5 VMEM per WMMA, 4x less L2 traffic
// on pooled).  Fragment layouts per CDNA5 ISA 7.12.2:
//   A (16x4 f32):  lane l holds M=l%16;  vgpr0 -> K = 2*(l/16), vgpr1 -> K+1
//   B (4x16 f32):  vgpr v, lane l holds  K = 2*v + (l/16), N = l%16
//   C/D (16x16):   vgpr v, lane l holds  M = v + 8*(l/16),  N = l%16
__global__ void __launch_bounds__(32)
gemm_kernel(const float* __restrict__ W,      // proj_w [256,256] row-major (e,d)
            const float* __restrict__ bias,   // proj_b [256]
            const float* __restrict__ pooled, // [64,256,512]
            float* __restrict__ out) {        // [64,256,511]
    const int lane = threadIdx.x;             // 0..31
    const int lrow = lane & 15;
    const int lhi  = lane >> 4;               // 0 or 1
    const int e0   = blockIdx.x * 64;         // 4 M tiles: rows e0 .. e0+63
    const int n0   = blockIdx.y * 16;         // N tile (padded N = 512)
    const int b    = blockIdx.z;

    const float2* W2 = (const float2*)W;
    const size_t wb0 = (size_t)(e0 +  0 + lrow) * (DDIM / 2);   // float2 units
    const size_t wb1 = (size_t)(e0 + 16 + lrow) * (DDIM / 2);
    const size_t wb2 = (size_t)(e0 + 32 + lrow) * (DDIM / 2);
    const size_t wb3 = (size_t)(e0 + 48 + lrow) * (DDIM / 2);
    const size_t pbase = ((size_t)b * DDIM) * NPAD + (size_t)(n0 + lrow);

    v8f c0 = {}, c1 = {}, c2 = {}, c3 = {};
#pragma unroll 2
    for (int k = 0; k < DDIM; k += 4) {
        const size_t wo = (size_t)(k >> 1) + lhi;
        float2 aw0 = W2[wb0 + wo];
        float2 aw1 = W2[wb1 + wo];
        float2 aw2 = W2[wb2 + wo];
        float2 aw3 = W2[wb3 + wo];

        const size_t pk = pbase + (size_t)(k + lhi) * NPAD;
        v2f bb;
        bb.x = pooled[pk];                    // row k+lhi
        bb.y = pooled[pk + 2 * NPAD];         // row k+2+lhi

        v2f a;
        a.x = aw0.x; a.y = aw0.y;
        c0 = __builtin_amdgcn_wmma_f32_16x16x4_f32(false, a, false, bb, (short)0, c0, false, false);
        a.x = aw1.x; a.y = aw1.y;
        c1 = __builtin_amdgcn_wmma_f32_16x16x4_f32(false, a, false, bb, (short)0, c1, false, false);
        a.x = aw2.x; a.y = aw2.y;
        c2 = __builtin_amdgcn_wmma_f32_16x16x4_f32(false, a, false, bb, (short)0, c2, false, false);
        a.x = aw3.x; a.y = aw3.y;
        c3 = __builtin_amdgcn_wmma_f32_16x16x4_f32(false, a, false, bb, (short)0, c3, false, false);
    }

    const int col = n0 + lrow;
    if (col < NWIN) {
        v8f acc[4] = { c0, c1, c2, c3 };
#pragma unroll
        for (int t = 0; t < 4; ++t) {
            const int erow = e0 + t * 16 + 8 * lhi;
            const size_t obase = ((size_t)b * DDIM + erow) * NWIN + col;
#pragma unroll
            for (int v = 0; v < 8; ++v) {
                float be = bias[erow + v];
                out[obase + (size_t)v * NWIN] = acc[t][v] + be;
            }
        }
    }
}

extern "C" void kernel_launch(void* const* d_in, const int* in_sizes, int n_in,
                              void* d_out, int out_size, void* d_ws, size_t ws_size,
                              hipStream_t stream) {
    const float* x    = (const float*)d_in[0];   // [64,256,4096]
    const float* pw   = (const float*)d_in[1];   // [3]
    const float* W    = (const float*)d_in[2];   // [256,256]
    const float* bias = (const float*)d_in[3];   // [256]
    float* out    = (float*)d_out;               // [64,256,511]
    float* pooled = (float*)d_ws;                // [64,256,512] = 32 MB scratch

    (void)in_sizes; (void)n_in; (void)out_size; (void)ws_size;

    pool_kernel<<<dim3(BATCH * DDIM), 256, 0, stream>>>(x, pw, pooled);
    gemm_kernel<<<dim3(DDIM / 64, NPAD / 16, BATCH), 32, 0, stream>>>(W, bias, pooled, out);
}